// Kalman_8315056685391
// MI455X (gfx1250) — compile-verified
//
#include <hip/hip_runtime.h>

// Kalman recursion on gfx1250 (MI455X), single persistent workgroup
// (1024 threads = 32 wave32s on one WGP).
//  - fp32 GEMMs on V_WMMA_F32_16X16X4_F32 with 2x2 register blocking
//  - K-panels staged LDS <- global by the Tensor Data Mover
//    (tensor_load_to_lds + s_wait_tensorcnt), double-buffered
//  - 64x64 innovation inverse via Gauss-Jordan in LDS (aliased onto panels)

typedef __attribute__((ext_vector_type(2))) float        v2f;
typedef __attribute__((ext_vector_type(8))) float        v8f;
typedef __attribute__((ext_vector_type(4))) unsigned int v4u;
typedef __attribute__((ext_vector_type(8))) int          v8i;
typedef __attribute__((ext_vector_type(4))) int          v4i;

#define TPB    1024
#define NWAVES 32
#define H      256
#define D      64
#define PANE   4096          // floats per panel buffer (max M*16 = 256*16)

// Issue one TDM 2D tile load: global (tile_d1 rows x tile_d0 contiguous f32,
// row stride `stride0` elements) -> LDS at byte offset lds_byte (packed).
__device__ __forceinline__ void tdm_load_2d(unsigned lds_byte, const void* gptr,
                                            unsigned tile_d0, unsigned tile_d1,
                                            unsigned stride0)
{
    unsigned long long ga = (unsigned long long)(uintptr_t)gptr;
    v4u g0;
    g0.x = 1u;                                          // count=1, user mode
    g0.y = lds_byte;                                    // lds_addr [63:32]
    g0.z = (unsigned)(ga & 0xffffffffu);                // global_addr [95:64]
    g0.w = (unsigned)((ga >> 32) & 0x01ffffffu)         // global_addr [120:96]
         | (2u << 30);                                  // type=2 (image)
    v8i g1;
    g1[0] = (int)(2u << 16);                            // data_size=2 -> 4 bytes
    g1[1] = (int)((tile_d0 & 0xffffu) << 16);           // tensor_dim0 lo16
    g1[2] = (int)((tile_d0 >> 16) & 0xffffu)            // tensor_dim0 hi16
          | (int)((tile_d1 & 0xffffu) << 16);           // tensor_dim1 lo16
    g1[3] = (int)((tile_d1 >> 16) & 0xffffu)            // tensor_dim1 hi16
          | (int)((tile_d0 & 0xffffu) << 16);           // tile_dim0
    g1[4] = (int)(tile_d1 & 0xffffu);                   // tile_dim1 (tile_dim2=0)
    g1[5] = (int)stride0;                               // tensor_dim0_stride lo32
    g1[6] = 0;                                          // stride0 hi / stride1 lo
    g1[7] = 0;
    v4i z4 = {0, 0, 0, 0};
    v8i z8 = {0, 0, 0, 0, 0, 0, 0, 0};
    __builtin_amdgcn_tensor_load_to_lds(g0, g1, z4, z4, z8, 0);
}

__device__ __forceinline__ unsigned lds_off(const void* p)
{
    return (unsigned)(uintptr_t)p;   // flat addr low 32 bits == LDS byte offset
}

// C[MxN] = scale * (A[MxK] @ B(^T)) {+ addm | addm - .}   addMode: 0/1/2
// Panels double-buffered in LDS, fetched by the TDM from wave 0.
// Each wave owns up to two 32x32 output blocks (2x2 of 16x16 WMMA tiles).
__device__ __forceinline__ void wg_gemm(const float* __restrict__ Am,
                                        const float* __restrict__ Bm,
                                        float* __restrict__ Cm,
                                        const float* __restrict__ addm,
                                        float scale, int addMode,
                                        int M, int N, int K,
                                        int lda, int ldb, int ldc,
                                        bool transB, int tid,
                                        float* shA, float* shB)
{
    const int lane = tid & 31;
    const int wave = tid >> 5;
    const int l15  = lane & 15;
    const int half = lane >> 4;
    const int h2   = half << 1;
    const int nWTn = N >> 5;
    const int nWT  = (M >> 5) * nWTn;
    const int nKB  = K >> 4;

    int  wm[2], wn[2];
    bool valid[2];
#pragma unroll
    for (int j = 0; j < 2; ++j) {
        const int wt = wave + j * NWAVES;
        valid[j] = wt < nWT;
        wm[j] = valid[j] ? (wt / nWTn) << 5 : 0;
        wn[j] = valid[j] ? (wt % nWTn) << 5 : 0;
    }

    v8f acc[2][2][2];
#pragma unroll
    for (int j = 0; j < 2; ++j)
#pragma unroll
        for (int mi = 0; mi < 2; ++mi)
#pragma unroll
            for (int ni = 0; ni < 2; ++ni)
                acc[j][mi][ni] = (v8f){0.f,0.f,0.f,0.f,0.f,0.f,0.f,0.f};

    // ---- prologue: TDM fetch of panel 0 ----
    if (wave == 0) {
        tdm_load_2d(lds_off(shA), Am, 16u, (unsigned)M, (unsigned)lda);
        if (transB)
            tdm_load_2d(lds_off(shB), Bm, 16u, (unsigned)N, (unsigned)ldb);
        else
            tdm_load_2d(lds_off(shB), Bm, (unsigned)N, 16u, (unsigned)ldb);
        __builtin_amdgcn_s_wait_tensorcnt((short)0);
    }
    __syncthreads();

    for (int kb = 0; kb < nKB; ++kb) {
        const int cur = kb & 1;
        if (wave == 0 && kb + 1 < nKB) {   // prefetch next panel into other buf
            const int k0 = (kb + 1) << 4;
            tdm_load_2d(lds_off(shA + (cur ^ 1) * PANE), Am + k0,
                        16u, (unsigned)M, (unsigned)lda);
            if (transB)
                tdm_load_2d(lds_off(shB + (cur ^ 1) * PANE), Bm + k0,
                            16u, (unsigned)N, (unsigned)ldb);
            else
                tdm_load_2d(lds_off(shB + (cur ^ 1) * PANE),
                            Bm + (size_t)k0 * ldb,
                            (unsigned)N, 16u, (unsigned)ldb);
        }
        const float* pA = shA + cur * PANE;
        const float* pB = shB + cur * PANE;

#pragma unroll
        for (int j = 0; j < 2; ++j) {
            if (!valid[j]) continue;
            const int a0r = wm[j] + l15;
            const int a1r = a0r + 16;
            const int c0  = wn[j] + l15;
            const int c1  = c0 + 16;
#pragma unroll
            for (int kl = 0; kl < 16; kl += 4) {
                const int kk = kl + h2;
                v2f a0 = *(const v2f*)(pA + a0r * 16 + kk);
                v2f a1 = *(const v2f*)(pA + a1r * 16 + kk);
                v2f b0, b1;
                if (transB) {
                    b0 = *(const v2f*)(pB + c0 * 16 + kk);
                    b1 = *(const v2f*)(pB + c1 * 16 + kk);
                } else {
                    b0.x = pB[kk * N + c0];  b0.y = pB[(kk + 1) * N + c0];
                    b1.x = pB[kk * N + c1];  b1.y = pB[(kk + 1) * N + c1];
                }
                acc[j][0][0] = __builtin_amdgcn_wmma_f32_16x16x4_f32(
                    false, a0, false, b0, (short)0, acc[j][0][0], false, false);
                acc[j][0][1] = __builtin_amdgcn_wmma_f32_16x16x4_f32(
                    false, a0, false, b1, (short)0, acc[j][0][1], false, false);
                acc[j][1][0] = __builtin_amdgcn_wmma_f32_16x16x4_f32(
                    false, a1, false, b0, (short)0, acc[j][1][0], false, false);
                acc[j][1][1] = __builtin_amdgcn_wmma_f32_16x16x4_f32(
                    false, a1, false, b1, (short)0, acc[j][1][1], false, false);
            }
        }
        if (wave == 0 && kb + 1 < nKB)
            __builtin_amdgcn_s_wait_tensorcnt((short)0);
        __syncthreads();
    }

    // ---- epilogue: scaled store with optional add / reverse-subtract ----
#pragma unroll
    for (int j = 0; j < 2; ++j) {
        if (!valid[j]) continue;
#pragma unroll
        for (int mi = 0; mi < 2; ++mi)
#pragma unroll
            for (int ni = 0; ni < 2; ++ni) {
                const int row0 = wm[j] + (mi << 4) + (half << 3);
                const int col  = wn[j] + (ni << 4) + l15;
#pragma unroll
                for (int r = 0; r < 8; ++r) {
                    const size_t idx = (size_t)(row0 + r) * ldc + col;
                    float v = acc[j][mi][ni][r] * scale;
                    if (addMode == 1)      v += addm[idx];
                    else if (addMode == 2) v  = addm[idx] - v;
                    Cm[idx] = v;
                }
            }
    }
}

__global__ void __launch_bounds__(TPB)
kalman_kernel(const float* __restrict__ X,
              const unsigned char* __restrict__ mask,
              const float* __restrict__ A,
              const float* __restrict__ Bm,
              const float* __restrict__ Q,
              const float* __restrict__ R,
              const float* __restrict__ s0,
              float* __restrict__ out,
              float* __restrict__ ws,
              int T)
{
    const int tid = threadIdx.x;

    // L2-resident workspace (floats).
    float* cov  = ws;               // H*H posterior covariance
    float* P    = cov  + H * H;     // H*H prior covariance
    float* AC   = P    + H * H;     // H*H A @ cov
    float* BP   = AC   + H * H;     // D*H B @ P
    float* S    = BP   + D * H;     // D*D innovation covariance
    float* Sinv = S    + D * D;     // D*D
    float* CB   = Sinv + D * D;     // H*D cov @ B^T
    float* Kg   = CB   + H * D;     // H*D Kalman gain
    float* KB   = Kg   + H * D;     // H*H K @ B

    __shared__ float smem[4 * PANE];    // panels; GJ scratch aliases onto it
    __shared__ float sh_state[H];
    __shared__ float sh_prior[H];
    __shared__ float sh_innov[D];
    __shared__ float sh_fac[D];
    float* shA = smem;                  // 2 * PANE
    float* shB = smem + 2 * PANE;       // 2 * PANE
    float (*shS)[2 * D] = (float (*)[2 * D])smem;   // 64 x 128 GJ augmented

    for (int i = tid; i < H * H; i += TPB)
        cov[i] = ((i / H) == (i % H)) ? 1.f : 0.f;
    if (tid < H) sh_state[tid] = s0[tid];
    __syncthreads();

    for (int t = 0; t < T; ++t) {
        // 1) AC = A @ cov
        wg_gemm(A, cov, AC, nullptr, 1.f, 0, H, H, H, H, H, H, false, tid, shA, shB);
        __syncthreads();
        // 2) P = AC @ A^T + Q
        wg_gemm(AC, A, P, Q, 1.f, 1, H, H, H, H, H, H, true, tid, shA, shB);
        __syncthreads();
        // 3) BP = B @ P
        wg_gemm(Bm, P, BP, nullptr, 1.f, 0, D, H, H, H, H, H, false, tid, shA, shB);
        __syncthreads();
        // 4) S = BP @ B^T + R
        wg_gemm(BP, Bm, S, R, 1.f, 1, D, D, H, H, H, D, true, tid, shA, shB);
        __syncthreads();

        // 5) Sinv = inv(S): Gauss-Jordan on [S | I] in LDS (aliases panels).
        for (int i = tid; i < D * D; i += TPB) {
            int r = i >> 6, c = i & 63;
            shS[r][c]     = S[i];
            shS[r][D + c] = (r == c) ? 1.f : 0.f;
        }
        __syncthreads();
        for (int p = 0; p < D; ++p) {
            const float pinv = 1.f / shS[p][p];
            __syncthreads();
            if (tid < 2 * D) shS[p][tid] *= pinv;
            __syncthreads();
            if (tid < D) sh_fac[tid] = shS[tid][p];
            __syncthreads();
            for (int i = tid; i < D * 2 * D; i += TPB) {
                int r = i >> 7, c = i & 127;
                if (r != p) shS[r][c] -= sh_fac[r] * shS[p][c];
            }
            __syncthreads();
        }
        for (int i = tid; i < D * D; i += TPB)
            Sinv[i] = shS[i >> 6][D + (i & 63)];
        __syncthreads();

        // 6) CB = cov @ B^T  (previous POSTERIOR cov, as in reference)
        wg_gemm(cov, Bm, CB, nullptr, 1.f, 0, H, D, H, H, H, D, true, tid, shA, shB);
        __syncthreads();
        // 7) K = (CB @ Sinv) * mask[t]   (zero-gain fold)
        const float mval = mask[t] ? 1.f : 0.f;
        wg_gemm(CB, Sinv, Kg, nullptr, mval, 0, H, D, D, D, D, D, false, tid, shA, shB);
        __syncthreads();
        // 8) KB = K @ B
        wg_gemm(Kg, Bm, KB, nullptr, 1.f, 0, H, H, D, D, H, H, false, tid, shA, shB);
        __syncthreads();
        // 9) cov = P - KB @ P   (reverse-subtract epilogue)
        wg_gemm(KB, P, cov, P, 1.f, 2, H, H, H, H, H, H, false, tid, shA, shB);
        __syncthreads();

        // 10) state recursion (cheap matvecs)
        if (tid < H) {
            float s = 0.f;
            const float* ar = A + (size_t)tid * H;
            for (int k = 0; k < H; ++k) s += ar[k] * sh_state[k];
            sh_prior[tid] = s;
        }
        __syncthreads();
        if (tid < D) {
            float s = 0.f;
            const float* br = Bm + (size_t)tid * H;
            for (int k = 0; k < H; ++k) s += br[k] * sh_prior[k];
            sh_innov[tid] = X[(size_t)t * D + tid] - s;
        }
        __syncthreads();
        if (tid < H) {
            float s = sh_prior[tid];
            const float* kr = Kg + (size_t)tid * D;
            for (int j = 0; j < D; ++j) s += kr[j] * sh_innov[j];
            sh_state[tid] = s;
            out[(size_t)t * H + tid] = s;
        }
        __syncthreads();
    }
}

extern "C" void kernel_launch(void* const* d_in, const int* in_sizes, int n_in,
                              void* d_out, int out_size, void* d_ws, size_t ws_size,
                              hipStream_t stream)
{
    // setup_inputs order: X, timeline, mask, A, B, Q, R, s_0
    const float*         X    = (const float*)d_in[0];
    const unsigned char* mask = (const unsigned char*)d_in[2];
    const float*         A    = (const float*)d_in[3];
    const float*         B    = (const float*)d_in[4];
    const float*         Q    = (const float*)d_in[5];
    const float*         R    = (const float*)d_in[6];
    const float*         s0   = (const float*)d_in[7];
    const int T = in_sizes[2];   // 2048

    kalman_kernel<<<1, TPB, 0, stream>>>(X, mask, A, B, Q, R, s0,
                                         (float*)d_out, (float*)d_ws, T);
}